// CCA_89266600280350
// MI455X (gfx1250) — compile-verified
//
#include <hip/hip_runtime.h>
#include <math.h>

// ---------------- problem constants ----------------
#define BATCH   8
#define IC      512
#define HW      4096          // 64*64
#define NMAP    (BATCH*HW)    // 32768 conv-output elements
#define ROWLEN4 (HW/4)        // 1024 float4 per (b,c) row
#define NROWS   (BATCH*IC)    // 4096 rows of length HW

// ---------------- workspace layout (in floats) ----------------
#define WS_M      0           // 32768 : raw conv output m[b*4096+j]
#define WS_X      32768       // 32768 : normalized X[b*4096+j]
#define WS_SCALE  65536       //  4096 : per-position final scale (1 + detal*s1[j])
#define WS_PSUM   69632       //   256 : per-block partial sums of m
#define WS_PSQ    69888       //   256 : per-block partial sums of m^2
#define WS_STATS  70144       //     2 : mu, rsig
#define WS_VPART  70152       //   128 : 16 blocks x 8 partial sums of X over j

typedef __attribute__((ext_vector_type(2))) float v2f;
typedef __attribute__((ext_vector_type(8))) float v8f;

// =====================================================================
// K1: 1x1 conv via V_WMMA_F32_16X16X4_F32.
// grid = 256 blocks x 256 threads (8 waves). Each wave -> 16 positions,
// reduced over 512 channels in 128 WMMA steps of K=4.
//   A (16x4)  : weight chunk replicated on all 16 M rows
//   B (4x16)  : x[b, c0..c0+3, j0..j0+15]
// fp32 A 16x4 layout : lanes 0-15 hold K0(v0),K1(v1); lanes 16-31 K2,K3
// fp32 B 4x16 layout : N = lane&15;   same K split by lane half
// => per-lane channel offset koff = 2*(lane>>4) works for BOTH A and B.
// All D rows identical, so acc[0] = m[j0 + (lane&15)] on every lane.
// =====================================================================
__global__ void __launch_bounds__(256)
k1_conv_wmma(const float* __restrict__ x, const float* __restrict__ conv_w,
             const float* __restrict__ conv_b, float* __restrict__ ws) {
    __shared__ float lw[IC];
    __shared__ float red[256];

    const int t = threadIdx.x;
    lw[t]       = conv_w[t];
    lw[t + 256] = conv_w[t + 256];
    __syncthreads();

    const int wave = t >> 5;
    const int lane = t & 31;
    const int half = lane >> 4;      // 0: K0/K1, 1: K2/K3
    const int l    = lane & 15;
    const int koff = 2 * half;

    const int P0 = blockIdx.x * 128 + wave * 16;   // global position base
    const int b  = P0 >> 12;                       // batch (16-chunk never crosses)
    const int j0 = P0 & 4095;                      // spatial index base

    const float* xb = x + (size_t)b * (IC * HW) + (size_t)koff * HW + j0 + l;

    v8f acc = {};
    #pragma unroll 4
    for (int c0 = 0; c0 < IC; c0 += 4) {
        v2f a, bm;
        a[0]  = lw[c0 + koff];
        a[1]  = lw[c0 + koff + 1];
        bm[0] = xb[(size_t)c0 * HW];
        bm[1] = xb[(size_t)c0 * HW + HW];
        acc = __builtin_amdgcn_wmma_f32_16x16x4_f32(
            /*neg_a=*/false, a, /*neg_b=*/false, bm,
            /*c_mod=*/(short)0, acc, /*reuse_a=*/false, /*reuse_b=*/false);
    }

    const float val = acc[0] + conv_b[0];
    if (lane < 16) ws[WS_M + b * HW + j0 + l] = val;

    // deterministic block tree-reduction of sum / sumsq (lanes<16 only contribute)
    const float s = (lane < 16) ? val : 0.0f;
    red[t] = s;
    __syncthreads();
    for (int o = 128; o > 0; o >>= 1) {
        if (t < o) red[t] += red[t + o];
        __syncthreads();
    }
    float Stot = 0.0f;
    if (t == 0) Stot = red[0];
    __syncthreads();

    red[t] = s * s;
    __syncthreads();
    for (int o = 128; o > 0; o >>= 1) {
        if (t < o) red[t] += red[t + o];
        __syncthreads();
    }
    if (t == 0) {
        ws[WS_PSUM + blockIdx.x] = Stot;
        ws[WS_PSQ  + blockIdx.x] = red[0];
    }
}

// =====================================================================
// K2: finalize BN batch statistics (single block, deterministic)
// =====================================================================
__global__ void __launch_bounds__(256)
k2_stats(float* __restrict__ ws) {
    __shared__ float r1[256];
    __shared__ float r2[256];
    const int t = threadIdx.x;
    r1[t] = ws[WS_PSUM + t];
    r2[t] = ws[WS_PSQ  + t];
    __syncthreads();
    for (int o = 128; o > 0; o >>= 1) {
        if (t < o) { r1[t] += r1[t + o]; r2[t] += r2[t + o]; }
        __syncthreads();
    }
    if (t == 0) {
        const float inv_n = 1.0f / (float)NMAP;
        const float mu  = r1[0] * inv_n;
        const float var = r2[0] * inv_n - mu * mu;     // biased variance
        ws[WS_STATS]     = mu;
        ws[WS_STATS + 1] = 1.0f / sqrtf(var + 1e-5f);
    }
}

// =====================================================================
// K3: per-position BN + PReLU + batch-dim normalization -> X[b,j];
// partial sums of X over j for v[b]. grid = 16 x 256 (one thread per j)
// =====================================================================
__global__ void __launch_bounds__(256)
k3_normalize(const float* __restrict__ bn_gamma, const float* __restrict__ bn_beta,
             const float* __restrict__ prelu_w, float* __restrict__ ws) {
    __shared__ float red[256];
    const int t = threadIdx.x;
    const int j = blockIdx.x * 256 + t;

    const float mu   = ws[WS_STATS];
    const float rsig = ws[WS_STATS + 1];
    const float gm = bn_gamma[0], bt = bn_beta[0], pw = prelu_w[0];

    float mn[BATCH];
    float n2 = 0.0f;
    #pragma unroll
    for (int b = 0; b < BATCH; ++b) {
        float m = ws[WS_M + b * HW + j];
        float v = (m - mu) * rsig * gm + bt;
        v = (v > 0.0f) ? v : pw * v;
        mn[b] = v;
        n2 += v * v;
    }
    const float nrm = sqrtf(n2);

    float Xl[BATCH];
    #pragma unroll
    for (int b = 0; b < BATCH; ++b) {
        Xl[b] = mn[b] / nrm;
        ws[WS_X + b * HW + j] = Xl[b];
    }

    // per-block partial sum over j for each batch row (deterministic)
    for (int b = 0; b < BATCH; ++b) {
        red[t] = Xl[b];
        __syncthreads();
        for (int o = 128; o > 0; o >>= 1) {
            if (t < o) red[t] += red[t + o];
            __syncthreads();
        }
        if (t == 0) ws[WS_VPART + blockIdx.x * 8 + b] = red[0];
        __syncthreads();
    }
}

// =====================================================================
// K4: v[b] = (1/64) * sum_j X[b,j]; then per position j:
//   g[j]  = 50/64 - 2*(X[:,j].v - 64)
//   s1[j] = 1/64 - 0.01*g/(|g|+1e-8)
//   scale[j] = 1 + detal*s1[j]
// single block (tiny work)
// =====================================================================
__global__ void __launch_bounds__(256)
k4_scale(const float* __restrict__ detal, float* __restrict__ ws) {
    __shared__ float v[BATCH];
    const int t = threadIdx.x;
    if (t < BATCH) {
        float s = 0.0f;
        for (int i = 0; i < 16; ++i) s += ws[WS_VPART + i * 8 + t];
        v[t] = s * (1.0f / 64.0f);
    }
    __syncthreads();

    const float dt = detal[0];
    for (int j = t; j < HW; j += 256) {
        float dot = 0.0f;
        #pragma unroll
        for (int b = 0; b < BATCH; ++b) dot += ws[WS_X + b * HW + j] * v[b];
        const float g  = (50.0f / 64.0f) - 2.0f * (dot - 64.0f);
        const float s1 = (1.0f / 64.0f) - 0.01f * g / (fabsf(g) + 1e-8f);
        ws[WS_SCALE + j] = 1.0f + dt * s1;
    }
}

// =====================================================================
// K5: out[b,c,j] = x[b,c,j] * scale[j].  One block per (b,c) row of 4096
// floats, float4-vectorized. Dominant HBM traffic (64 MiB R + 64 MiB W).
// =====================================================================
__global__ void __launch_bounds__(256)
k5_apply(const float* __restrict__ x, const float* __restrict__ ws,
         float* __restrict__ out) {
    const int t = threadIdx.x;
    const size_t base = (size_t)blockIdx.x * ROWLEN4;
    const float4* x4 = (const float4*)x;
    const float4* s4 = (const float4*)(ws + WS_SCALE);
    float4* o4 = (float4*)out;
    #pragma unroll
    for (int u = 0; u < 4; ++u) {
        const int p = u * 256 + t;              // 0..1023 within the row
        const float4 xv = x4[base + p];
        const float4 sv = s4[p];
        float4 r;
        r.x = xv.x * sv.x;
        r.y = xv.y * sv.y;
        r.z = xv.z * sv.z;
        r.w = xv.w * sv.w;
        o4[base + p] = r;
    }
}

// =====================================================================
extern "C" void kernel_launch(void* const* d_in, const int* in_sizes, int n_in,
                              void* d_out, int out_size, void* d_ws, size_t ws_size,
                              hipStream_t stream) {
    (void)in_sizes; (void)n_in; (void)out_size; (void)ws_size;
    const float* x        = (const float*)d_in[0];
    const float* conv_w   = (const float*)d_in[1];
    const float* conv_b   = (const float*)d_in[2];
    const float* bn_gamma = (const float*)d_in[3];
    const float* bn_beta  = (const float*)d_in[4];
    const float* prelu_w  = (const float*)d_in[5];
    const float* detal    = (const float*)d_in[6];
    float* ws  = (float*)d_ws;
    float* out = (float*)d_out;

    k1_conv_wmma<<<256, 256, 0, stream>>>(x, conv_w, conv_b, ws);
    k2_stats    <<<1,   256, 0, stream>>>(ws);
    k3_normalize<<<16,  256, 0, stream>>>(bn_gamma, bn_beta, prelu_w, ws);
    k4_scale    <<<1,   256, 0, stream>>>(detal, ws);
    k5_apply    <<<NROWS, 256, 0, stream>>>(x, ws, out);
}